// PDE_kernel_34394098106732
// MI455X (gfx1250) — compile-verified
//
#include <hip/hip_runtime.h>
#include <math.h>

typedef float v2f __attribute__((ext_vector_type(2)));
typedef float v8f __attribute__((ext_vector_type(8)));

#define B_N 8
#define V_N 13
#define H_N 32
#define W_N 64
#define CIN 256
#define VD 65
#define PLANE 2048              /* 32*64 */
#define FLD (B_N*V_N*PLANE)     /* 212992 */
#define NH (B_N*VD*PLANE)       /* 1064960 */
#define NOUT (B_N*CIN*PLANE)    /* 4194304 */

#define PI_F 3.14159265358979f
#define RADIUS_F 6371000.0f
#define PY_F (PI_F*RADIUS_F/33.0f)
#define OMEGA_F 7.29e-5f
#define LV_F 2.5e6f
#define RGAS_F 8.314f
#define CP_F 1005.0f
#define RD_F 287.0f
#define DT_F 300.0f
#define KH_F 15.0f
#define KV_F 0.1f
#define SIGMA_F 5.67e-8f
#define EMIS_F 0.7f
#define PTH_F 0.8f
#define DR_F 0.05f

__constant__ float c_dz[13]   = {50,50,50,50,50,75,100,100,100,125,112,75,75};
__constant__ float c_pres[13] = {50,100,150,200,250,300,400,500,600,700,850,925,1000};

// ---------------- device helpers ----------------
__device__ __forceinline__ float latf(int y) {
    return (90.0f - (float)(y + 1) * (180.0f / 33.0f)) * (PI_F / 180.0f);
}
__device__ __forceinline__ float pixx(int y) {
    return 2.0f * PI_F * RADIUS_F * cosf(latf(y)) / 64.0f;
}
__device__ __forceinline__ float avoid_inf(float t) {
    if (fabsf(t) == 0.0f) t = 0.1f;
    if (fabsf(t) < 1.0f) t = copysignf(1.0f, t);
    return t;
}
__device__ __forceinline__ int pyid(int i) { return i < 2 ? i : (i < 34 ? i - 2 : i - 4); }
__device__ __forceinline__ int pzid(int i) { return i < 2 ? i : (i < 15 ? i - 2 : i - 4); }

__device__ __forceinline__ float d_x_at(const float* f, int b, int l, int y, int x) {
    const float* p = f + ((b * V_N + l) * H_N + y) * W_N;
    float g = (p[(x + 62) & 63] - 8.0f * p[(x + 63) & 63]
             + 8.0f * p[(x + 1) & 63] - p[(x + 2) & 63]) * (1.0f / 12.0f);
    return g / pixx(y);
}
__device__ __forceinline__ float d_y_at(const float* f, int b, int l, int y, int x) {
    const float* base = f + (b * V_N + l) * PLANE + x;
    float a = base[pyid(y) * 64], bb = base[pyid(y + 1) * 64];
    float c = base[pyid(y + 3) * 64], d = base[pyid(y + 4) * 64];
    return (-a + 8.0f * bb - 8.0f * c + d) * (1.0f / 12.0f) / PY_F;
}
__device__ __forceinline__ float d_z_at(const float* f, int b, int l, int y, int x) {
    const float* base = f + b * V_N * PLANE + y * 64 + x;
    float a = base[pzid(l) * PLANE], bb = base[pzid(l + 1) * PLANE];
    float c = base[pzid(l + 3) * PLANE], d = base[pzid(l + 4) * PLANE];
    return (-a + 8.0f * bb - 8.0f * c + d) * (1.0f / 12.0f) / c_dz[l];
}
__device__ __forceinline__ float mixing_at(const float* f, int b, int l, int y, int x) {
    // d_x(d_x f): PIXEL_X constant along x -> composite 9-tap kernel / px^2
    const float k2[9] = {1.f, -16.f, 64.f, 16.f, -130.f, 16.f, 64.f, -16.f, 1.f};
    const float* p = f + ((b * V_N + l) * H_N + y) * W_N;
    float s = 0.0f;
#pragma unroll
    for (int o = -4; o <= 4; ++o) s += k2[o + 4] * p[(x + o + 64) & 63];
    float px = pixx(y);
    float dxx = s * (1.0f / 144.0f) / (px * px);
    float dyy = (-d_y_at(f, b, l, pyid(y), x) + 8.0f * d_y_at(f, b, l, pyid(y + 1), x)
               - 8.0f * d_y_at(f, b, l, pyid(y + 3), x) + d_y_at(f, b, l, pyid(y + 4), x))
              * (1.0f / 12.0f) / PY_F;
    float dzz = (-d_z_at(f, b, pzid(l), y, x) + 8.0f * d_z_at(f, b, pzid(l + 1), y, x)
               - 8.0f * d_z_at(f, b, pzid(l + 3), y, x) + d_z_at(f, b, pzid(l + 4), y, x))
              * (1.0f / 12.0f) / c_dz[l];
    return KH_F * (dxx + dyy) + KV_F * dzz;
}

// ---------------- weight packing (A-fragments, lane-major pairs) ----------------
__global__ void pack_w1(const float* __restrict__ w, float* __restrict__ wp) {
    int i = blockIdx.x * 256 + threadIdx.x;
    if (i >= 9 * 64 * 5 * 64) return;
    int lanevg = i & 63;
    int t = i >> 6;
    int mt = t % 5; t /= 5;
    int icq = t % 64; int p = t / 64;
    int vg = lanevg & 1, lane = lanevg >> 1;
    int half = lane >> 4;
    int kk = vg + 2 * half;             // A layout: VGPR0 holds K=0/2, VGPR1 holds K=1/3
    int oc = mt * 16 + (lane & 15);
    int ic = icq * 4 + kk;
    int ky = p / 3, kx = p % 3;
    wp[i] = (oc < VD) ? w[((oc * CIN + ic) * 3 + ky) * 3 + kx] : 0.0f;
}

__global__ void pack_w2(const float* __restrict__ w, float* __restrict__ wp) {
    int i = blockIdx.x * 256 + threadIdx.x;
    if (i >= 9 * 17 * 16 * 64) return;
    int lanevg = i & 63;
    int t = i >> 6;
    int mt = t % 16; t /= 16;
    int icq = t % 17; int p = t / 17;
    int vg = lanevg & 1, lane = lanevg >> 1;
    int half = lane >> 4;
    int kk = vg + 2 * half;
    int oc = mt * 16 + (lane & 15);
    int ic = icq * 4 + kk;
    int ky = p / 3, kx = p % 3;
    wp[i] = (ic < VD) ? w[((oc * VD + ic) * 3 + ky) * 3 + kx] : 0.0f;
}

// ---------------- conv1: [8,256,32,64] -> [8,65,32,64], fp32 WMMA 16x16x4 ----------------
__global__ __launch_bounds__(32) void conv1_wmma(const float* __restrict__ x,
                                                 const float* __restrict__ wpack,
                                                 const float* __restrict__ bias,
                                                 float* __restrict__ h) {
    int tile = blockIdx.x;                 // 1024 N-tiles of 16 columns
    int lane = threadIdx.x;
    int b = tile >> 7, rem = tile & 127;
    int y = rem >> 2, x0 = (rem & 3) << 4;
    int col = lane & 15, half = lane >> 4;
    int xcol = x0 + col;
    v8f acc[5] = {};
    const float* xb = x + (size_t)b * CIN * PLANE;
    for (int p = 0; p < 9; ++p) {
        int ky = p / 3, kx = p % 3;
        int ys = y + ky - 1;
        int xs = xcol + kx - 1;
        bool inb = (ys >= 0) && (ys < H_N) && (xs >= 0) && (xs < W_N);
        // Unconditional loads from a clamped (always-valid) address; zero-padding
        // applied as a mask multiply -> no EXEC branching in the hot loop.
        float mask = inb ? 1.0f : 0.0f;
        const float* xrow = xb + (inb ? (ys * 64 + xs) : 0);
        const float* wp = wpack + p * (64 * 5 * 64);
#pragma unroll 2
        for (int icq = 0; icq < 64; ++icq) {
            int ic0 = icq * 4 + 2 * half;  // B layout mirrors A: VGPR0 K=0/2, VGPR1 K=1/3
            v2f bf;
            bf[0] = xrow[(size_t)ic0 * PLANE] * mask;
            bf[1] = xrow[(size_t)(ic0 + 1) * PLANE] * mask;
            const float* wq = wp + icq * (5 * 64);
#pragma unroll
            for (int m = 0; m < 5; ++m) {
                v2f af = *(const v2f*)(wq + m * 64 + lane * 2);
                acc[m] = __builtin_amdgcn_wmma_f32_16x16x4_f32(
                    false, af, false, bf, (short)0, acc[m], false, false);
            }
        }
    }
#pragma unroll
    for (int m = 0; m < 5; ++m)
#pragma unroll
        for (int g = 0; g < 8; ++g) {
            int oc = m * 16 + g + 8 * half;   // D layout: VGPR g = rows g / g+8
            if (oc < VD)
                h[((b * VD + oc) * H_N + y) * W_N + xcol] = acc[m][g] + bias[oc];
        }
}

// ---------------- conv2: [8,65,32,64] -> [8,256,32,64] ----------------
__global__ __launch_bounds__(128) void conv2_wmma(const float* __restrict__ mix,
                                                  const float* __restrict__ wpack,
                                                  const float* __restrict__ bias,
                                                  float* __restrict__ out) {
    int tile = blockIdx.x;                 // 1024 N-tiles
    int wave = threadIdx.x >> 5;
    int lane = threadIdx.x & 31;
    int b = tile >> 7, rem = tile & 127;
    int y = rem >> 2, x0 = (rem & 3) << 4;
    int col = lane & 15, half = lane >> 4;
    int xcol = x0 + col;
    v8f acc[4] = {};
    const float* mb = mix + (size_t)b * VD * PLANE;
    for (int p = 0; p < 9; ++p) {
        int ky = p / 3, kx = p % 3;
        int ys = y + ky - 1;
        int xs = xcol + kx - 1;
        bool inb = (ys >= 0) && (ys < H_N) && (xs >= 0) && (xs < W_N);
        float smask = inb ? 1.0f : 0.0f;
        const float* mrow = mb + (inb ? (ys * 64 + xs) : 0);
        const float* wp = wpack + p * (17 * 16 * 64);
#pragma unroll 2
        for (int icq = 0; icq < 17; ++icq) {
            int ic0 = icq * 4 + 2 * half;
            // clamp channel index (stays inside this batch's 65-channel block),
            // fold the ic<65 tail guard into the mask multiply
            int ic0c = ic0 < VD - 1 ? ic0 : VD - 1;
            int ic1c = ic0 + 1 < VD - 1 ? ic0 + 1 : VD - 1;
            float m0 = (ic0 < VD) ? smask : 0.0f;
            float m1 = (ic0 + 1 < VD) ? smask : 0.0f;
            v2f bf;
            bf[0] = mrow[(size_t)ic0c * PLANE] * m0;
            bf[1] = mrow[(size_t)ic1c * PLANE] * m1;
            const float* wq = wp + (icq * 16 + wave * 4) * 64;
#pragma unroll
            for (int m = 0; m < 4; ++m) {
                v2f af = *(const v2f*)(wq + m * 64 + lane * 2);
                acc[m] = __builtin_amdgcn_wmma_f32_16x16x4_f32(
                    false, af, false, bf, (short)0, acc[m], false, false);
            }
        }
    }
#pragma unroll
    for (int m = 0; m < 4; ++m)
#pragma unroll
        for (int g = 0; g < 8; ++g) {
            int oc = (wave * 4 + m) * 16 + g + 8 * half;
            out[((size_t)(b * CIN + oc) * H_N + y) * W_N + xcol] = acc[m][g] + bias[oc];
        }
}

// ---------------- batchnorm statistics (per channel over B,H,W) ----------------
__global__ void bn_stats(const float* __restrict__ x, int C, float* __restrict__ stats) {
    int c = blockIdx.x, t = threadIdx.x;
    float s = 0.0f, s2 = 0.0f;
    for (int i = t; i < B_N * PLANE; i += 256) {
        int b = i >> 11, sp = i & 2047;
        float v = x[((size_t)b * C + c) * PLANE + sp];
        s += v; s2 += v * v;
    }
    __shared__ float ls[256], ls2[256];
    ls[t] = s; ls2[t] = s2; __syncthreads();
    for (int o = 128; o > 0; o >>= 1) {
        if (t < o) { ls[t] += ls[t + o]; ls2[t] += ls2[t + o]; }
        __syncthreads();
    }
    if (t == 0) {
        float m = ls[0] * (1.0f / 16384.0f);
        stats[c * 2] = m;
        stats[c * 2 + 1] = ls2[0] * (1.0f / 16384.0f) - m * m;
    }
}

// ---------------- BN apply + split into 5 fields ----------------
__global__ void bn_split(const float* __restrict__ h, const float* __restrict__ stats,
                         const float* gz, const float* bz, const float* gq, const float* bq,
                         const float* gu, const float* bu, const float* gv, const float* bv,
                         const float* gt, const float* bt,
                         float* Z, float* Q, float* U, float* Vf, float* T) {
    int i = blockIdx.x * 256 + threadIdx.x;
    if (i >= NH) return;
    int sp = i & 2047;
    int c = (i >> 11) % VD;
    int b = (i >> 11) / VD;
    int f = c / 13, cc = c % 13;
    const float *g, *bb; float* dst;
    if (f == 0)      { g = gz; bb = bz; dst = Z; }
    else if (f == 1) { g = gq; bb = bq; dst = Q; }
    else if (f == 2) { g = gu; bb = bu; dst = U; }
    else if (f == 3) { g = gv; bb = bv; dst = Vf; }
    else             { g = gt; bb = bt; dst = T; }
    float v = (h[i] - stats[c * 2]) * rsqrtf(stats[c * 2 + 1] + 1e-5f) * g[cc] + bb[cc];
    dst[(b * V_N + cc) * PLANE + sp] = v;
}

// ---------------- fixed derivative fields ----------------
__global__ void deriv_fields(const float* Z, const float* U, const float* Vf,
                             float* ZX, float* ZY, float* ZZ, float* UX, float* VY) {
    int i = blockIdx.x * 256 + threadIdx.x;
    if (i >= FLD) return;
    int x = i & 63, y = (i >> 6) & 31, bl = i >> 11, l = bl % 13, b = bl / 13;
    ZX[i] = d_x_at(Z, b, l, y, x);
    ZY[i] = d_y_at(Z, b, l, y, x);
    ZZ[i] = d_z_at(Z, b, l, y, x);
    UX[i] = d_x_at(U, b, l, y, x);
    VY[i] = d_y_at(Vf, b, l, y, x);
}

__global__ void wvel_k(const float* UX, const float* VY, float* WV) {
    int i = blockIdx.x * 256 + threadIdx.x;
    if (i >= B_N * PLANE) return;
    int sp = i & 2047, b = i >> 11;
    float s = 0.0f;
    for (int l = 0; l < V_N; ++l) {
        int id = (b * V_N + l) * PLANE + sp;
        s += c_dz[l] * (UX[id] + VY[id]);
        WV[id] = -s;
    }
}

// ---------------- RK4 stages ----------------
__global__ void uv_stage(const float* cu_, const float* cv_, const float* ux_, const float* vy_,
                         const float* zx_, const float* zy_, const float* wv_,
                         float* ku_, float* kv_) {
    int i = blockIdx.x * 256 + threadIdx.x;
    if (i >= FLD) return;
    int x = i & 63, y = (i >> 6) & 31, bl = i >> 11, l = bl % 13, b = bl / 13;
    float cu = cu_[i], cv = cv_[i], w = wv_[i];
    float fcor = 2.0f * OMEGA_F * sinf(latf(y));
    ku_[i] = -cu * ux_[i] - cv * d_y_at(cu_, b, l, y, x) - w * d_z_at(cu_, b, l, y, x)
             + fcor * cv - zx_[i] + mixing_at(cu_, b, l, y, x);
    kv_[i] = -cu * d_x_at(cv_, b, l, y, x) - cv * vy_[i] - w * d_z_at(cv_, b, l, y, x)
             - fcor * cu - zy_[i] + mixing_at(cv_, b, l, y, x);
}

__global__ void t_stage(const float* ct_, const float* u_, const float* v_, const float* wv_,
                        const float* q_, const float* qs_, const float* zz_, float* kt_) {
    int i = blockIdx.x * 256 + threadIdx.x;
    if (i >= FLD) return;
    int x = i & 63, y = (i >> 6) & 31, bl = i >> 11, l = bl % 13, b = bl / 13;
    float w = wv_[i], zz = zz_[i], t = ct_[i];
    float q = q_[i], qs = qs_[i];
    float rate = (q / avoid_inf(qs) > PTH_F) ? (q - PTH_F * qs) * (1.0f / DT_F) : 0.0f;
    float lh = rate * (LV_F / CP_F);
    float Qh = -LV_F * zz * w;
    float ta = t + 273.15f;
    float ta2 = ta * ta;
    float rc = -(EMIS_F * SIGMA_F * ta2 * ta2) * (RD_F * ta) / (CP_F * c_pres[l] * 100.0f);
    kt_[i] = (Qh - zz * w) * (1.0f / CP_F)
             - u_[i] * d_x_at(ct_, b, l, y, x) - v_[i] * d_y_at(ct_, b, l, y, x)
             - w * d_z_at(ct_, b, l, y, x) + mixing_at(ct_, b, l, y, x) + rc + lh;
}

__global__ void rk_accum(const float* __restrict__ k, float* __restrict__ sum,
                         const float* __restrict__ base, float* __restrict__ cur,
                         float wt, float alpha, int first, int n) {
    int i = blockIdx.x * 256 + threadIdx.x;
    if (i >= n) return;
    float kv = k[i];
    sum[i] = first ? (wt * kv) : (sum[i] + wt * kv);
    if (alpha != 0.0f) cur[i] = base[i] + alpha * DT_F * kv;
}

// ---------------- saturation path ----------------
__global__ void earg_k(const float* T, float* E) {
    int i = blockIdx.x * 256 + threadIdx.x;
    if (i >= FLD) return;
    float tc = T[i] - 273.15f;
    E[i] = 17.67f * tc / avoid_inf(tc + 243.5f);
}

__global__ void qs_k(const float* T, const float* ZZ, const float* est, float* QS) {
    int i = blockIdx.x * 256 + threadIdx.x;
    if (i >= FLD) return;
    float t = T[i];
    float tc = t - 273.15f;
    float e = 17.67f * tc / avoid_inf(tc + 243.5f);
    float emn = est[0], emx = est[1];
    float sc = (e - emn) / (emx - emn) * (3.01f + 3.47f) - 3.47f;
    float es = 6.112f * expf(sc) * 100.0f;
    float rho = -1.0f / avoid_inf(ZZ[i]);
    float p = rho * RGAS_F * t;
    QS[i] = fmaxf(0.622f * es / avoid_inf(p - 0.378f * es), 1e-6f);
}

// ---------------- z and q tendencies ----------------
__global__ void ztend_k(const float* KT, float* ZT) {
    int i = blockIdx.x * 256 + threadIdx.x;
    if (i >= B_N * PLANE) return;
    int sp = i & 2047, b = i >> 11;
    float s = 0.0f;
    for (int l = 0; l < V_N; ++l) {
        int id = (b * V_N + l) * PLANE + sp;
        s += c_dz[l] * (-RGAS_F / c_pres[l]) * KT[id];
        ZT[id] = s;
    }
}

__global__ void qtend_k(const float* Q, const float* Qs, const float* U, const float* Vf,
                        const float* WV, float* QT) {
    int i = blockIdx.x * 256 + threadIdx.x;
    if (i >= FLD) return;
    int x = i & 63, y = (i >> 6) & 31, bl = i >> 11, l = bl % 13, b = bl / 13;
    float q = Q[i], qs = Qs[i];
    float rate = (q / avoid_inf(qs) > PTH_F) ? (q - PTH_F * qs) * (1.0f / DT_F) : 0.0f;
    QT[i] = -U[i] * d_x_at(Q, b, l, y, x) - Vf[i] * d_y_at(Q, b, l, y, x)
            - WV[i] * d_z_at(Q, b, l, y, x) + mixing_at(Q, b, l, y, x) - rate;
}

// ---------------- reductions (deterministic two-pass) ----------------
__global__ void reduce_part(const float* __restrict__ f, int n, float* __restrict__ part) {
    int t = threadIdx.x;
    float mn = INFINITY, mx = -INFINITY, s = 0.0f;
    for (int i = blockIdx.x * 256 + t; i < n; i += 256 * 256) {
        float v = f[i];
        mn = fminf(mn, v); mx = fmaxf(mx, v); s += v;
    }
    __shared__ float smn[256], smx[256], ssm[256];
    smn[t] = mn; smx[t] = mx; ssm[t] = s; __syncthreads();
    for (int o = 128; o > 0; o >>= 1) {
        if (t < o) {
            smn[t] = fminf(smn[t], smn[t + o]);
            smx[t] = fmaxf(smx[t], smx[t + o]);
            ssm[t] += ssm[t + o];
        }
        __syncthreads();
    }
    if (t == 0) {
        part[blockIdx.x * 4 + 0] = smn[0];
        part[blockIdx.x * 4 + 1] = smx[0];
        part[blockIdx.x * 4 + 2] = ssm[0];
    }
}

__global__ void reduce_fin(const float* __restrict__ part, float inv_n, float* __restrict__ stat) {
    int t = threadIdx.x;
    __shared__ float smn[256], smx[256], ssm[256];
    smn[t] = part[t * 4 + 0]; smx[t] = part[t * 4 + 1]; ssm[t] = part[t * 4 + 2];
    __syncthreads();
    for (int o = 128; o > 0; o >>= 1) {
        if (t < o) {
            smn[t] = fminf(smn[t], smn[t + o]);
            smx[t] = fmaxf(smx[t], smx[t + o]);
            ssm[t] += ssm[t + o];
        }
        __syncthreads();
    }
    if (t == 0) { stat[0] = smn[0]; stat[1] = smx[0]; stat[2] = ssm[0] * inv_n; }
}

// ---------------- scale_diff update (in place) ----------------
__global__ void apply_sd(float* __restrict__ xf, const float* __restrict__ df,
                         const float* __restrict__ dstat, const float* __restrict__ xstat,
                         float ds, int n) {
    int i = blockIdx.x * 256 + threadIdx.x;
    if (i >= n) return;
    float dmn = dstat[0] * ds, dmx = dstat[1] * ds;
    float a = (xstat[0] - xstat[2]) * DR_F;
    float b = (xstat[1] - xstat[2]) * DR_F;
    float d = df[i] * ds;
    xf[i] += (d - dmn) / (dmx - dmn) * (b - a) + a;
}

// ---------------- blend phys with h ----------------
__global__ void mix_k(const float* __restrict__ h, const float* __restrict__ coef,
                      const float* Z, const float* Q, const float* U, const float* Vf,
                      const float* T, float* __restrict__ mo) {
    int i = blockIdx.x * 256 + threadIdx.x;
    if (i >= NH) return;
    int sp = i & 2047;
    int c = (i >> 11) % VD;
    int b = (i >> 11) / VD;
    int f = c / 13, cc = c % 13;
    const float* fld = (f == 0) ? Z : (f == 1) ? Q : (f == 2) ? U : (f == 3) ? Vf : T;
    float ph = fld[(b * V_N + cc) * PLANE + sp];
    float cf = coef[c * PLANE + sp];
    mo[i] = cf * ph + (1.0f - cf) * h[i];
}

// ---------------- final BN + residual ----------------
__global__ void bn_final(const float* __restrict__ c2o, const float* __restrict__ stats,
                         const float* gblk, const float* bblk, const float* __restrict__ x,
                         float* __restrict__ out) {
    int i = blockIdx.x * 256 + threadIdx.x;
    if (i >= NOUT) return;
    int c = (i >> 11) % CIN;
    out[i] = (c2o[i] - stats[c * 2]) * rsqrtf(stats[c * 2 + 1] + 1e-5f) * gblk[c] + bblk[c] + x[i];
}

// ---------------- host side ----------------
static void reduce3(const float* f, int n, float* part, float* slot, hipStream_t s) {
    reduce_part<<<256, 256, 0, s>>>(f, n, part);
    reduce_fin<<<1, 256, 0, s>>>(part, 1.0f / (float)n, slot);
}

extern "C" void kernel_launch(void* const* d_in, const int* in_sizes, int n_in,
                              void* d_out, int out_size, void* d_ws, size_t ws_size,
                              hipStream_t stream) {
    const float* X        = (const float*)d_in[0];
    const float* W1       = (const float*)d_in[1];
    const float* Bn1      = (const float*)d_in[2];
    const float* W2       = (const float*)d_in[3];
    const float* Bn2      = (const float*)d_in[4];
    const float* COEF     = (const float*)d_in[5];
    const float* gz = (const float*)d_in[6],  * bz = (const float*)d_in[7];
    const float* gq = (const float*)d_in[8],  * bq = (const float*)d_in[9];
    const float* gu = (const float*)d_in[10], * bu = (const float*)d_in[11];
    const float* gv = (const float*)d_in[12], * bv = (const float*)d_in[13];
    const float* gt = (const float*)d_in[14], * bt = (const float*)d_in[15];
    const float* gblk = (const float*)d_in[16], * bblk = (const float*)d_in[17];
    float* OUT = (float*)d_out;
    float* ws = (float*)d_ws;

    float* WP1 = ws;                       // 184320
    float* WP2 = WP1 + 184320;             // 156672
    float* Hb  = WP2 + 156672;             // NH
    float* MIX = Hb + NH;                  // NH
    float* C2O = MIX + NH;                 // NOUT
    float* F0  = C2O + NOUT;
    float* Z  = F0 + 0 * FLD; float* Q  = F0 + 1 * FLD; float* U  = F0 + 2 * FLD;
    float* Vf = F0 + 3 * FLD; float* T  = F0 + 4 * FLD;
    float* ZX = F0 + 5 * FLD; float* ZY = F0 + 6 * FLD; float* ZZ = F0 + 7 * FLD;
    float* UX = F0 + 8 * FLD; float* VY = F0 + 9 * FLD; float* WV = F0 + 10 * FLD;
    float* QS = F0 + 11 * FLD;
    float* CU = F0 + 12 * FLD; float* CV = F0 + 13 * FLD;
    float* KU = F0 + 14 * FLD; float* KV = F0 + 15 * FLD;
    float* SU = F0 + 16 * FLD; float* SV = F0 + 17 * FLD;
    float* CT = F0 + 18 * FLD; float* KT = F0 + 19 * FLD; float* STm = F0 + 20 * FLD;
    float* ZT = F0 + 21 * FLD; float* QT = F0 + 22 * FLD; float* TMP = F0 + 23 * FLD;
    float* ST1 = F0 + 24 * FLD;            // 130 BN1 stats
    float* ST2 = ST1 + 130;                // 512 BN2 stats
    float* GST = ST2 + 512;                // 12 slots * 4
    float* PART = GST + 48;                // 1024 partials

    int gF = (FLD + 255) / 256;
    int gH = (NH + 255) / 256;
    int gO = (NOUT + 255) / 256;
    int gP = (B_N * PLANE + 255) / 256;

    // 1. weight packing
    pack_w1<<<(9 * 64 * 5 * 64 + 255) / 256, 256, 0, stream>>>(W1, WP1);
    pack_w2<<<(9 * 17 * 16 * 64 + 255) / 256, 256, 0, stream>>>(W2, WP2);

    // 2. conv1 (WMMA f32) + BN + split
    conv1_wmma<<<1024, 32, 0, stream>>>(X, WP1, Bn1, Hb);
    bn_stats<<<VD, 256, 0, stream>>>(Hb, VD, ST1);
    bn_split<<<gH, 256, 0, stream>>>(Hb, ST1, gz, bz, gq, bq, gu, bu, gv, bv, gt, bt,
                                     Z, Q, U, Vf, T);

    // 3. fixed derivatives + vertical velocity
    deriv_fields<<<gF, 256, 0, stream>>>(Z, U, Vf, ZX, ZY, ZZ, UX, VY);
    wvel_k<<<gP, 256, 0, stream>>>(UX, VY, WV);

    // 4. base-field statistics for scale_diff
    reduce3(U, FLD, PART, GST + 0 * 4, stream);
    reduce3(Vf, FLD, PART, GST + 1 * 4, stream);
    reduce3(T, FLD, PART, GST + 2 * 4, stream);
    reduce3(Z, FLD, PART, GST + 3 * 4, stream);
    reduce3(Q, FLD, PART, GST + 4 * 4, stream);

    // 5. saturation q_s (needs global min/max of exponent argument)
    earg_k<<<gF, 256, 0, stream>>>(T, TMP);
    reduce3(TMP, FLD, PART, GST + 10 * 4, stream);
    qs_k<<<gF, 256, 0, stream>>>(T, ZZ, GST + 10 * 4, QS);

    // 6. RK4 momentum
    uv_stage<<<gF, 256, 0, stream>>>(U, Vf, UX, VY, ZX, ZY, WV, KU, KV);
    rk_accum<<<gF, 256, 0, stream>>>(KU, SU, U, CU, 1.0f, 0.5f, 1, FLD);
    rk_accum<<<gF, 256, 0, stream>>>(KV, SV, Vf, CV, 1.0f, 0.5f, 1, FLD);
    uv_stage<<<gF, 256, 0, stream>>>(CU, CV, UX, VY, ZX, ZY, WV, KU, KV);
    rk_accum<<<gF, 256, 0, stream>>>(KU, SU, U, CU, 2.0f, 0.5f, 0, FLD);
    rk_accum<<<gF, 256, 0, stream>>>(KV, SV, Vf, CV, 2.0f, 0.5f, 0, FLD);
    uv_stage<<<gF, 256, 0, stream>>>(CU, CV, UX, VY, ZX, ZY, WV, KU, KV);
    rk_accum<<<gF, 256, 0, stream>>>(KU, SU, U, CU, 2.0f, 1.0f, 0, FLD);
    rk_accum<<<gF, 256, 0, stream>>>(KV, SV, Vf, CV, 2.0f, 1.0f, 0, FLD);
    uv_stage<<<gF, 256, 0, stream>>>(CU, CV, UX, VY, ZX, ZY, WV, KU, KV);
    rk_accum<<<gF, 256, 0, stream>>>(KU, SU, U, CU, 1.0f, 0.0f, 0, FLD);
    rk_accum<<<gF, 256, 0, stream>>>(KV, SV, Vf, CV, 1.0f, 0.0f, 0, FLD);

    // 7. RK4 temperature (uses original u, v)
    t_stage<<<gF, 256, 0, stream>>>(T, U, Vf, WV, Q, QS, ZZ, KT);
    rk_accum<<<gF, 256, 0, stream>>>(KT, STm, T, CT, 1.0f, 0.5f, 1, FLD);
    t_stage<<<gF, 256, 0, stream>>>(CT, U, Vf, WV, Q, QS, ZZ, KT);
    rk_accum<<<gF, 256, 0, stream>>>(KT, STm, T, CT, 2.0f, 0.5f, 0, FLD);
    t_stage<<<gF, 256, 0, stream>>>(CT, U, Vf, WV, Q, QS, ZZ, KT);
    rk_accum<<<gF, 256, 0, stream>>>(KT, STm, T, CT, 2.0f, 1.0f, 0, FLD);
    t_stage<<<gF, 256, 0, stream>>>(CT, U, Vf, WV, Q, QS, ZZ, KT);
    rk_accum<<<gF, 256, 0, stream>>>(KT, STm, T, CT, 1.0f, 0.0f, 0, FLD);  // KT holds k4t

    // 8. geopotential and moisture tendencies (before in-place updates)
    ztend_k<<<gP, 256, 0, stream>>>(KT, ZT);
    qtend_k<<<gF, 256, 0, stream>>>(Q, QS, U, Vf, WV, QT);

    // 9. tendency statistics
    reduce3(SU, FLD, PART, GST + 5 * 4, stream);
    reduce3(SV, FLD, PART, GST + 6 * 4, stream);
    reduce3(STm, FLD, PART, GST + 7 * 4, stream);
    reduce3(ZT, FLD, PART, GST + 8 * 4, stream);
    reduce3(QT, FLD, PART, GST + 9 * 4, stream);

    // 10. scale_diff updates (in place)
    const float k6 = DT_F / 6.0f;
    apply_sd<<<gF, 256, 0, stream>>>(U, SU, GST + 5 * 4, GST + 0 * 4, k6, FLD);
    apply_sd<<<gF, 256, 0, stream>>>(Vf, SV, GST + 6 * 4, GST + 1 * 4, k6, FLD);
    apply_sd<<<gF, 256, 0, stream>>>(T, STm, GST + 7 * 4, GST + 2 * 4, k6, FLD);
    apply_sd<<<gF, 256, 0, stream>>>(Z, ZT, GST + 8 * 4, GST + 3 * 4, DT_F, FLD);
    apply_sd<<<gF, 256, 0, stream>>>(Q, QT, GST + 9 * 4, GST + 4 * 4, DT_F, FLD);

    // 11. blend, conv2 (WMMA f32), final BN + residual
    mix_k<<<gH, 256, 0, stream>>>(Hb, COEF, Z, Q, U, Vf, T, MIX);
    conv2_wmma<<<1024, 128, 0, stream>>>(MIX, WP2, Bn2, C2O);
    bn_stats<<<CIN, 256, 0, stream>>>(C2O, CIN, ST2);
    bn_final<<<gO, 256, 0, stream>>>(C2O, ST2, gblk, bblk, X, OUT);

    (void)in_sizes; (void)n_in; (void)out_size; (void)ws_size;
}